// DAS_49074296324299
// MI455X (gfx1250) — compile-verified
//
#include <hip/hip_runtime.h>

// ---- problem constants -----------------------------------------------------
#define S_    128
#define T_    10240
#define NX_   256
#define NY_   256
#define NPIX  (NX_ * NY_)
#define NCH   16            // B(8) * 2

typedef __attribute__((ext_vector_type(2))) float v2f;
typedef __attribute__((ext_vector_type(4))) float v4f;
typedef __attribute__((ext_vector_type(8))) float v8f;

__device__ constexpr float DXSQ     = 1.0e-6f;              // DX*DX (= DY*DY)
__device__ constexpr float INV_VSDT = 25806.451612903225f;  // 1/(VS*DT)

__device__ __forceinline__ v8f wmma_dis2(v2f a, v2f b, v8f c) {
  // D = A(16x4 f32) x B(4x16 f32) + C, full FP32; K=4 exactly fits the
  // rank-4 expansion of the squared sensor->pixel distance.
  return __builtin_amdgcn_wmma_f32_16x16x4_f32(false, a, false, b,
                                               (short)0, c, false, false);
}

// ---- pass 0: (ch, s, t) -> (s, t, ch) so one gather = 64 contiguous bytes --
__global__ __launch_bounds__(256) void k_transpose(const float* __restrict__ in,
                                                   float* __restrict__ outT) {
  const unsigned idx = blockIdx.x * blockDim.x + threadIdx.x;  // flat (s,t)
  float v[NCH];
#pragma unroll
  for (int c = 0; c < NCH; ++c)
    v[c] = in[(size_t)c * (size_t)(S_ * T_) + (size_t)idx];
  v4f* dst = (v4f*)(outT + ((size_t)idx << 4));
#pragma unroll
  for (int q = 0; q < 4; ++q) {
    v4f w;
    w.x = v[4 * q + 0]; w.y = v[4 * q + 1];
    w.z = v[4 * q + 2]; w.w = v[4 * q + 3];
    dst[q] = w;
  }
}

// ---- pass 1: WMMA distance + gather-accumulate -----------------------------
// One wave = 16 pixels (one row-tile). 8 WMMAs (one per 16-sensor group)
// produce dis^2 for 16 sensors x 16 pixels. Lane L handles pixel L%16;
// lanes L and L+16 each consume 8 sensor rows per group (C/D layout:
// VGPR r holds M=r for lanes 0-15 and M=r+8 for lanes 16-31).
template <bool TRANSPOSED>
__global__ __launch_bounds__(256) void k_das(const float* __restrict__ traceT,
                                             const float* __restrict__ data,
                                             const int*   __restrict__ mask,
                                             float*       __restrict__ out) {
  const int lane   = threadIdx.x & 31;
  const int waveId = blockIdx.x * (blockDim.x >> 5) + (threadIdx.x >> 5); // 0..4095
  const int i      = waveId >> 4;                 // pixel row 0..255
  const int n      = lane & 15;                   // pixel slot in tile (N col)
  const int jj     = ((waveId & 15) << 4) + n;    // pixel col
  const int half   = lane >> 4;                   // sensor-row half of D

  // B matrix (4x16): row0=1, row1=-2i, row2=-2j, row3=i^2+j^2
  // layout: VGPR0 = {K0 | K2}, VGPR1 = {K1 | K3} across lane halves.
  const float fi = (float)i, fj = (float)jj;
  v2f bmat;
  if (half == 0) { bmat.x = 1.0f;       bmat.y = -2.0f * fi;        }
  else           { bmat.x = -2.0f * fj; bmat.y = fi * fi + fj * fj; }

  float acc[NCH];
#pragma unroll
  for (int c = 0; c < NCH; ++c) acc[c] = 0.0f;

  for (int g = 0; g < 8; ++g) {
    // A matrix (16x4): row s = {(x^2+y^2), x, y, 1} * DX^2
    // layout (ISA table): lanes 0-15 -> K0,K1 ; lanes 16-31 -> K2,K3.
    const int   s = (g << 4) + n;
    const float x = (float)mask[2 * s + 0];
    const float y = (float)mask[2 * s + 1];
    v2f amat;
    if (half == 0) { amat.x = (x * x + y * y) * DXSQ; amat.y = x * DXSQ; }
    else           { amat.x = y * DXSQ;               amat.y = DXSQ;     }

    v8f czero = {0.f, 0.f, 0.f, 0.f, 0.f, 0.f, 0.f, 0.f};
    v8f d = wmma_dis2(amat, bmat, czero);   // dis^2 for 16 sensors x 16 px

    const int      sgBase   = (g << 4) + (half << 3);
    const unsigned gIdxBase = (unsigned)sgBase * (unsigned)T_; // hoisted s*T
#pragma unroll
    for (int r = 0; r < 8; ++r) {
      const float    d2 = d[r];
      // t >= 0 always; keep it unsigned so no sign-extend is emitted.
      const unsigned t  = (unsigned)(int)(__builtin_amdgcn_sqrtf(d2) * INV_VSDT);
      // 32-bit element index; whole trace buffer < 2^27 elements, so the
      // byte offset fits i32 -> saddr + zext(voffset) global loads.
      const unsigned eIdx = gIdxBase + (unsigned)(r * T_) + t;
      if (TRANSPOSED) {
        const unsigned byteOff = eIdx << 6;    // *16 channels *4 bytes
        const v4f* src = (const v4f*)((const char*)traceT + byteOff);
        v4f p0 = src[0], p1 = src[1], p2 = src[2], p3 = src[3];
        acc[0]  += p0.x; acc[1]  += p0.y; acc[2]  += p0.z; acc[3]  += p0.w;
        acc[4]  += p1.x; acc[5]  += p1.y; acc[6]  += p1.z; acc[7]  += p1.w;
        acc[8]  += p2.x; acc[9]  += p2.y; acc[10] += p2.z; acc[11] += p2.w;
        acc[12] += p3.x; acc[13] += p3.y; acc[14] += p3.z; acc[15] += p3.w;
      } else {
        const unsigned byteOff = eIdx << 2;
        const float* src = (const float*)((const char*)data + byteOff);
#pragma unroll
        for (int c = 0; c < NCH; ++c)
          acc[c] += src[(size_t)c * (size_t)(S_ * T_)];
      }
    }
  }

  // Lanes L and L+16 hold disjoint 64-sensor partial sums of the same pixel.
#pragma unroll
  for (int c = 0; c < NCH; ++c)
    acc[c] += __shfl_xor(acc[c], 16, 32);

  if (half == 0) {
    const size_t p = (size_t)i * NY_ + (size_t)jj;
#pragma unroll
    for (int c = 0; c < NCH; ++c)
      out[(size_t)c * NPIX + p] = acc[c];   // coalesced over 16 lanes
  }
}

// ---- pass 2: per-channel min/max + in-place normalize ----------------------
__global__ __launch_bounds__(256) void k_norm(float* __restrict__ img) {
  const int c = blockIdx.x;                 // one block per channel
  float* base = img + (size_t)c * NPIX;

  __shared__ float smn[8];
  __shared__ float smx[8];
  __shared__ float fin[2];

  float mn = 3.0e38f, mx = -3.0e38f;
  for (int p = threadIdx.x; p < NPIX; p += 256) {
    const float v = base[p];
    mn = fminf(mn, v);
    mx = fmaxf(mx, v);
  }
#pragma unroll
  for (int off = 16; off >= 1; off >>= 1) {
    mn = fminf(mn, __shfl_xor(mn, off, 32));
    mx = fmaxf(mx, __shfl_xor(mx, off, 32));
  }
  const int wv = threadIdx.x >> 5;
  if ((threadIdx.x & 31) == 0) { smn[wv] = mn; smx[wv] = mx; }
  __syncthreads();
  if (threadIdx.x == 0) {
    float a = smn[0], b = smx[0];
#pragma unroll
    for (int w = 1; w < 8; ++w) { a = fminf(a, smn[w]); b = fmaxf(b, smx[w]); }
    fin[0] = a; fin[1] = b;
  }
  __syncthreads();
  const float a   = fin[0];
  const float inv = 1.0f / (fin[1] - a);
  for (int p = threadIdx.x; p < NPIX; p += 256)
    base[p] = (base[p] - a) * inv;
}

// ---- host glue -------------------------------------------------------------
extern "C" void kernel_launch(void* const* d_in, const int* in_sizes, int n_in,
                              void* d_out, int out_size, void* d_ws, size_t ws_size,
                              hipStream_t stream) {
  const float* data = (const float*)d_in[0];  // (B,2,S,T) f32
  const int*   mask = (const int*)  d_in[1];  // (S,2) i32
  float*       out  = (float*)d_out;          // (B,2,NX,NY) f32

  const size_t needT = (size_t)S_ * T_ * NCH * sizeof(float);  // ~84 MB
  const dim3 dasGrid(NPIX / (16 * 8));   // 4096 wave-tiles / 8 waves per block
  const dim3 dasBlk(256);

  if (ws_size >= needT) {
    float* traceT = (float*)d_ws;
    k_transpose<<<(S_ * T_) / 256, 256, 0, stream>>>(data, traceT);
    k_das<true><<<dasGrid, dasBlk, 0, stream>>>(traceT, data, mask, out);
  } else {
    k_das<false><<<dasGrid, dasBlk, 0, stream>>>(nullptr, data, mask, out);
  }
  k_norm<<<NCH, 256, 0, stream>>>(out);
  (void)in_sizes; (void)n_in; (void)out_size;
}